// Gemma3nConformerSDPA_2173253452007
// MI455X (gfx1250) — compile-verified
//
#include <hip/hip_runtime.h>
#include <hip/hip_bf16.h>
#include <math.h>

// ---- problem constants (match reference) ----
#define B_SZ   16
#define T_LEN  2040
#define HEADS  8
#define HD     192           // head dim
#define MODEL  1536
#define W_CH   12            // CHUNK
#define CTXC   24            // CTX
#define F_SPAN 13
#define UBLK   170           // T / CHUNK

typedef float v2f __attribute__((ext_vector_type(2)));
typedef float v8f __attribute__((ext_vector_type(8)));

// ---------------------------------------------------------------------------
// Prep kernel: dscale[k] = (HD^-0.5 / log 2) * softplus(per_dim_scale[k])
//              sheads[h][f][k] = (sin_emb @ pos_proj_w^T)[f, h*HD + k]
// ---------------------------------------------------------------------------
__global__ void conformer_prep_kernel(const float* __restrict__ pos_proj_w,
                                      const float* __restrict__ per_dim_scale,
                                      float* __restrict__ dscale,
                                      float* __restrict__ sheads) {
    int idx = blockIdx.x * blockDim.x + threadIdx.x;
    if (idx < HD) {
        float x  = per_dim_scale[idx];
        float sp = (x > 20.f) ? x : log1pf(__expf(x));
        dscale[idx] = (rsqrtf((float)HD) / 0.6931471805599453f) * sp;
    }
    if (idx >= F_SPAN * MODEL) return;
    int f = idx / MODEL;
    int d = idx % MODEL;
    float p = (float)(12 - f);                 // pos = 12..0
    const float inc = 9.210340371976184f / 767.f;   // log(1e4)/(n-1), n=768
    const float* wrow = pos_proj_w + (size_t)d * MODEL;
    float acc = 0.f;
    for (int n = 0; n < MODEL / 2; ++n) {
        float ts = __expf(-(float)n * inc);
        float s, c;
        __sincosf(p * ts, &s, &c);
        acc += s * wrow[n] + c * wrow[MODEL / 2 + n];
    }
    int h = d / HD, kk = d % HD;
    sheads[((size_t)h * F_SPAN + f) * HD + kk] = acc;
}

// ---------------------------------------------------------------------------
// Main kernel: one wave32 per (b, u, h)
// ---------------------------------------------------------------------------
__launch_bounds__(32)
__global__ void conformer_attn_kernel(const float* __restrict__ q,
                                      const float* __restrict__ k,
                                      const float* __restrict__ v,
                                      const unsigned char* __restrict__ mask,
                                      const float* __restrict__ dscale,
                                      const float* __restrict__ sheads,
                                      float* __restrict__ out) {
    __shared__ __align__(16) float sQ [16   * HD];    // scaled Q, rows 12-15 zero
    __shared__ __align__(16) float sK [CTXC * HD];    // key context, row-major
    __shared__ __align__(16) float sVt[HD * CTXC];    // V transposed: [feat][c]
    __shared__ __align__(16) float sS [F_SPAN * HD];  // pos-proj rows for this head
    __shared__ __align__(16) float sL [16 * 32];      // AC logits
    __shared__ __align__(16) float sBd[16 * 16];      // BD term
    __shared__ __align__(16) float sP [16 * 32];      // probs (cols 24-31, rows>=12 zero)

    const int tid = threadIdx.x;
    const int bid = blockIdx.x;
    const int h = bid % HEADS;
    const int u = (bid / HEADS) % UBLK;
    const int b = bid / (HEADS * UBLK);
    const int t0 = u * W_CH;

    // ---- stage Q (scaled) ----
    for (int i = tid; i < W_CH * HD; i += 32) {
        int w = i / HD, kk = i % HD;
        sQ[i] = q[(((size_t)b * T_LEN + t0 + w) * HEADS + h) * HD + kk] * dscale[kk];
    }
    for (int i = tid; i < 4 * HD; i += 32) sQ[W_CH * HD + i] = 0.f;
    // ---- stage K and V^T (zero-padded where t out of range) ----
    for (int i = tid; i < CTXC * HD; i += 32) {
        int c = i / HD, kk = i % HD;
        int t = t0 + c - 12;
        bool ok = (t >= 0) && (t < T_LEN);
        size_t gi = (((size_t)b * T_LEN + (ok ? t : 0)) * HEADS + h) * HD + kk;
        sK [c * HD + kk]   = ok ? k[gi] : 0.f;
        sVt[kk * CTXC + c] = ok ? v[gi] : 0.f;
    }
    // ---- stage positional projection for this head ----
    for (int i = tid; i < F_SPAN * HD; i += 32)
        sS[i] = sheads[(size_t)h * F_SPAN * HD + i];
    for (int i = tid; i < 16 * 32; i += 32) sP[i] = 0.f;
    __syncthreads();

    const int half = tid >> 4;       // 0: K pair (0,1); 1: K pair (2,3)
    const int lm   = tid & 15;
    const int koff = 2 * half;
    const int c1r  = (16 + lm < CTXC) ? (16 + lm) : (CTXC - 1); // clamp; cols>=24 discarded
    const int frow = (lm < F_SPAN) ? lm : (F_SPAN - 1);          // clamp; cols>=13 discarded

    // ---- AC (two 16-wide N tiles) and BD, fused over K=192 ----
    v8f acc0 = {}; v8f acc1 = {}; v8f accS = {};
    for (int kk = 0; kk < HD; kk += 4) {
        v2f a  = *(const v2f*)&sQ[lm  * HD + kk + koff];
        v2f b0 = *(const v2f*)&sK[lm  * HD + kk + koff];
        v2f b1 = *(const v2f*)&sK[c1r * HD + kk + koff];
        v2f bs = *(const v2f*)&sS[frow * HD + kk + koff];
        acc0 = __builtin_amdgcn_wmma_f32_16x16x4_f32(false, a, false, b0, (short)0, acc0, false, false);
        acc1 = __builtin_amdgcn_wmma_f32_16x16x4_f32(false, a, false, b1, (short)0, acc1, false, false);
        accS = __builtin_amdgcn_wmma_f32_16x16x4_f32(false, a, false, bs, (short)0, accS, false, false);
    }
    // D layout: VGPR r -> M = r + 8*half, N = lm
    for (int r = 0; r < 8; ++r) {
        int m = r + 8 * half;
        sL [m * 32 + lm]      = acc0[r];
        sL [m * 32 + 16 + lm] = acc1[r];
        sBd[m * 16 + lm]      = accS[r];
    }
    __syncthreads();

    // ---- relative shift + tanh cap + mask + softmax (one lane per query row) ----
    if (tid < W_CH) {
        const int w = tid;
        float vals[CTXC];
        float mx = -3.0e38f;
        for (int c = 0; c < CTXC; ++c) {
            int f = c - w;
            bool causal = (f >= 0) && (f <= F_SPAN - 1);
            float bd = causal ? sBd[w * 16 + f] : 0.f;
            float lg = sL[w * 32 + c] + bd;
            lg = 50.f * tanhf(lg * 0.02f);
            int t = t0 + c - 12;
            bool tin = (t >= 0) && (t < T_LEN);
            bool ok = causal && tin && (mask[(size_t)b * T_LEN + (tin ? t : 0)] == 0);
            lg = ok ? lg : -3.0e38f;
            vals[c] = lg;
            mx = fmaxf(mx, lg);
        }
        float ssum = 0.f;
        for (int c = 0; c < CTXC; ++c) { float e = __expf(vals[c] - mx); ssum += e; vals[c] = e; }
        float inv = 1.f / ssum;
        for (int c = 0; c < CTXC; ++c) sP[w * 32 + c] = vals[c] * inv;
    }
    __syncthreads();

    // ---- PV: P(16x24 padded to x32) x V(24x192), 12 N-tiles ----
    for (int nt = 0; nt < HD / 16; ++nt) {
        v8f acc = {};
        const int vc = nt * 16 + lm;            // output feature column for this lane
        for (int kk = 0; kk < CTXC; kk += 4) {
            v2f a  = *(const v2f*)&sP [lm * 32 + kk + koff];
            v2f bb = *(const v2f*)&sVt[vc * CTXC + kk + koff];
            acc = __builtin_amdgcn_wmma_f32_16x16x4_f32(false, a, false, bb, (short)0, acc, false, false);
        }
        for (int r = 0; r < 8; ++r) {
            int m = r + 8 * half;
            if (m < W_CH) {
                out[(((size_t)b * T_LEN + t0 + m) * HEADS + h) * HD + nt * 16 + lm] = acc[r];
            }
        }
    }
}

// ---------------------------------------------------------------------------
extern "C" void kernel_launch(void* const* d_in, const int* in_sizes, int n_in,
                              void* d_out, int out_size, void* d_ws, size_t ws_size,
                              hipStream_t stream) {
    const float*         q    = (const float*)d_in[0];
    const float*         k    = (const float*)d_in[1];
    const float*         v    = (const float*)d_in[2];
    const unsigned char* mask = (const unsigned char*)d_in[3];   // jnp.bool_ -> 1 byte
    const float*         ppw  = (const float*)d_in[4];
    const float*         pds  = (const float*)d_in[5];

    float* dscale = (float*)d_ws;                 // 192 floats
    float* sheads = dscale + HD;                  // 8*13*192 floats

    int prep_threads = F_SPAN * MODEL;            // 19968 (also covers dscale's 192)
    conformer_prep_kernel<<<(prep_threads + 255) / 256, 256, 0, stream>>>(ppw, pds, dscale, sheads);

    conformer_attn_kernel<<<B_SZ * UBLK * HEADS, 32, 0, stream>>>(
        q, k, v, mask, dscale, sheads, (float*)d_out);
}